// ResonantLayer_11965778886785
// MI455X (gfx1250) — compile-verified
//
#include <hip/hip_runtime.h>
#include <hip/hip_bf16.h>
#include <math.h>
#include <stdint.h>

#define D_MODEL 1024
#define NUM_NEURONS 4096
#define BATCH 16
#define PHI_F 1.61803398874989484820f

typedef __attribute__((ext_vector_type(4))) unsigned int uint4v;
typedef __attribute__((ext_vector_type(8))) unsigned int uint8v;
typedef __attribute__((ext_vector_type(2))) float v2f;
typedef __attribute__((ext_vector_type(8))) float v8f;

union F4 { float4 v; float f[4]; };
union F2 { float2 v; float f[2]; };

// ---------------------------------------------------------------------------
// Kernel 0: drive, stored TRANSPOSED: drvT[d*16 + b]
//   drive[b,d] = exp(ma[d]) * (xr[b,d]*cos(ang[d]+t[b]) + xi[b,d]*sin(ang[d]+t[b]))
// ---------------------------------------------------------------------------
__global__ __launch_bounds__(256) void drive_kernel(
    const float* __restrict__ xr, const float* __restrict__ xi,
    const float* __restrict__ t,  const float* __restrict__ ma,
    const float* __restrict__ ang, float* __restrict__ drvT) {
  int idx = blockIdx.x * 256 + threadIdx.x;        // 0 .. 16383
  int b = idx & 15;
  int d = idx >> 4;
  float total = ang[d] + t[b];
  float c = cosf(total);
  float s = sinf(total);
  int src = b * D_MODEL + d;
  drvT[idx] = expf(ma[d]) * (xr[src] * c + xi[src] * s);
}

// ---------------------------------------------------------------------------
// Kernel 1: resonant layer. 512 blocks x 256 threads (8 waves), one neuron
// per wave. drvT (64KB) staged to LDS by one TDM descriptor; per-neuron
// (wl,phase) table built in LDS; d-reduction done on the matrix pipe with
// V_WMMA_F32_16X16X4_F32 (A = trig*attn products, B = ones).
// ---------------------------------------------------------------------------
__global__ __launch_bounds__(256) void resonant_kernel(
    const float* __restrict__ W,  const float* __restrict__ Bm,
    const float* __restrict__ AC, const float* __restrict__ AS,
    const float* __restrict__ drive, float* __restrict__ out) {
  __shared__ float drvT[D_MODEL * BATCH];          // 64 KB, layout [d][b]
  __shared__ float2 wlph[8][D_MODEL];              // 64 KB, per-wave (wl, phase)

  const int lane = threadIdx.x & 31;
  const int wave = threadIdx.x >> 5;               // 0..7
  const int n = blockIdx.x * 8 + wave;             // neuron row

  // --- issue TDM: wave 0 loads all of drvT (64KB) into LDS ---------------
  if (threadIdx.x < 32) {
    uint32_t lds_off = (uint32_t)(uintptr_t)(&drvT[0]);
    uint64_t ga = (uint64_t)(uintptr_t)drive;
    const uint32_t NTOT = BATCH * D_MODEL;         // 16384 f32 elements

    uint4v g0;
    g0[0] = 1u;                                    // count=1
    g0[1] = lds_off;                               // lds_addr
    g0[2] = (uint32_t)ga;                          // global_addr lo
    g0[3] = (uint32_t)(ga >> 32) | (2u << 30);     // global_addr hi | type=2

    uint8v g1;
    g1[0] = (2u << 16);                            // data_size = 4 bytes
    g1[1] = (NTOT & 0xFFFFu) << 16;                // tensor_dim0[15:0]
    g1[2] = (NTOT >> 16) | (1u << 16);             // tensor_dim0 hi | tensor_dim1=1
    g1[3] = (NTOT << 16);                          // tile_dim0 = 16384
    g1[4] = 1u;                                    // tile_dim1=1, tile_dim2=0
    g1[5] = NTOT;                                  // tensor_dim0_stride
    g1[6] = 0u;
    g1[7] = 0u;
    asm volatile("tensor_load_to_lds %0, %1" :: "s"(g0), "s"(g1) : "memory");
  }

  // --- prologue (overlaps TDM): per-(n,d) wl = rcp(1+|W|), phase = B+PHI*|W|
  const float* __restrict__ Wr  = W  + (size_t)n * D_MODEL;
  const float* __restrict__ Br  = Bm + (size_t)n * D_MODEL;
  const float* __restrict__ ACr = AC + (size_t)n * D_MODEL;
  const float* __restrict__ ASr = AS + (size_t)n * D_MODEL;

#pragma unroll
  for (int i = 0; i < 8; ++i) {
    const int d4 = (i * 32 + lane) * 4;
    F4 w4, b4;
    w4.v = *(const float4*)(Wr + d4);              // global_load_b128
    b4.v = *(const float4*)(Br + d4);
    F4 lo, hi;
#pragma unroll
    for (int j = 0; j < 4; ++j) {
      float aw = fabsf(w4.f[j]);
      float wl = __builtin_amdgcn_rcpf(1.0f + aw); // v_rcp_f32, once per (n,d)
      float ph = fmaf(PHI_F, aw, b4.f[j]);
      if (j < 2) { lo.f[2*j] = wl; lo.f[2*j+1] = ph; }
      else       { hi.f[2*(j-2)] = wl; hi.f[2*(j-2)+1] = ph; }
    }
    *(float4*)(&wlph[wave][d4])     = lo.v;        // ds_store_b128
    *(float4*)(&wlph[wave][d4 + 2]) = hi.v;
  }

  if (threadIdx.x < 32) __builtin_amdgcn_s_wait_tensorcnt(0);
  __syncthreads();

  // --- main loop: lane = (half h, batch b); WMMA reduces over d -----------
  const int b    = lane & 15;                      // A-matrix row M
  const int h    = lane >> 4;                      // 0 -> K=0,1   1 -> K=2,3
  const int dofs = h << 1;                         // dA = 4c + 2h

  const v2f ones = {1.0f, 1.0f};                   // B matrix: 4x16 all-ones
  v8f accC = {0.f, 0.f, 0.f, 0.f, 0.f, 0.f, 0.f, 0.f};
  v8f accS = {0.f, 0.f, 0.f, 0.f, 0.f, 0.f, 0.f, 0.f};

#pragma unroll 4
  for (int c4 = 0; c4 < D_MODEL; c4 += 4) {
    const int dA = c4 + dofs;

    F4 wp;                                         // {wl0, ph0, wl1, ph1}
    wp.v = *(const float4*)(&wlph[wave][dA]);      // ds_load_b128 (broadcast)
    float dv0 = drvT[dA * BATCH + b];              // ds_load_b32, conflict-free
    float dv1 = drvT[(dA + 1) * BATCH + b];
    F2 acp, asp;
    acp.v = *(const float2*)(ACr + dA);            // global b64 (broadcast)
    asp.v = *(const float2*)(ASr + dA);

    float th0 = fmaf(dv0, wp.f[0], wp.f[1]);
    float th1 = fmaf(dv1, wp.f[2], wp.f[3]);
    float c0 = __cosf(th0), s0 = __sinf(th0);      // v_cos_f32 / v_sin_f32
    float c1 = __cosf(th1), s1 = __sinf(th1);

    v2f aC = {c0 * acp.f[0], c1 * acp.f[1]};       // A[m=b, k] = cos*attn
    v2f aS = {s0 * asp.f[0], s1 * asp.f[1]};

    accC = __builtin_amdgcn_wmma_f32_16x16x4_f32(
        false, aC, false, ones, (short)0, accC, false, false);
    accS = __builtin_amdgcn_wmma_f32_16x16x4_f32(
        false, aS, false, ones, (short)0, accS, false, false);
  }

  // --- writeback straight from C-matrix layout ----------------------------
  // C layout: vgpr r, lanes 0-15: row M=r; lanes 16-31: row M=8+r.
  // All 16 columns are identical (B = ones), so every lane holds 8 totals.
  const int r = lane & 7;
  float vc = accC[0], vs = accS[0];
#pragma unroll
  for (int q = 1; q < 8; ++q) {
    vc = (r == q) ? accC[q] : vc;
    vs = (r == q) ? accS[q] : vs;
  }
  const bool isSin = (lane & 8) != 0;              // lanes 8-15 / 24-31 -> sin
  const int brow = (h << 3) + r;                   // batch index 0..15
  const float val = (isSin ? vs : vc) * 0.03125f;  // 1/sqrt(1024)
  const size_t base = isSin ? (size_t)BATCH * NUM_NEURONS : 0;
  out[base + (size_t)brow * NUM_NEURONS + n] = val;
}

// ---------------------------------------------------------------------------
extern "C" void kernel_launch(void* const* d_in, const int* in_sizes, int n_in,
                              void* d_out, int out_size, void* d_ws, size_t ws_size,
                              hipStream_t stream) {
  const float* xr  = (const float*)d_in[0];   // x_real        (16,1024)
  const float* xi  = (const float*)d_in[1];   // x_imag        (16,1024)
  const float* t   = (const float*)d_in[2];   // t             (16,)
  const float* ma  = (const float*)d_in[3];   // measure_amplitude (1024,)
  const float* ang = (const float*)d_in[4];   // measure_angle (1024,)
  const float* W   = (const float*)d_in[5];   // W             (4096,1024)
  const float* Bm  = (const float*)d_in[6];   // B             (4096,1024)
  const float* ac  = (const float*)d_in[7];   // attn_cos      (4096,1024)
  const float* as  = (const float*)d_in[8];   // attn_sin      (4096,1024)

  float* drvT = (float*)d_ws;                 // 16*1024 f32 = 64KB, [d][b]
  float* out  = (float*)d_out;                // [cos (16,4096) | sin (16,4096)]

  drive_kernel<<<dim3(64), dim3(256), 0, stream>>>(xr, xi, t, ma, ang, drvT);
  resonant_kernel<<<dim3(512), dim3(256), 0, stream>>>(W, Bm, ac, as, drvT, out);
}